// Upsampling_Conv_55886114456171
// MI455X (gfx1250) — compile-verified
//
#include <hip/hip_runtime.h>
#include <hip/hip_bf16.h>

#define WW 2048
#define HH 64
#define NB 32
#define PQSTRIDE 2050             // p/q arrays: halo at 0 and 2049, data at 1..2048

typedef int v4i __attribute__((vector_size(4 * sizeof(int))));

#if __has_builtin(__builtin_amdgcn_global_load_async_to_lds_b128)
#define HAVE_ASYNC_LOAD 1
#else
#define HAVE_ASYNC_LOAD 0
#endif
#if __has_builtin(__builtin_amdgcn_global_store_async_from_lds_b128)
#define HAVE_ASYNC_STORE 1
#else
#define HAVE_ASYNC_STORE 0
#endif

__device__ __forceinline__ void wait_async0() {
#if __has_builtin(__builtin_amdgcn_s_wait_asynccnt)
    __builtin_amdgcn_s_wait_asynccnt(0);
#else
    asm volatile("s_wait_asynccnt 0" ::: "memory");
#endif
}

// ---------------- Pass 1: per-block min partials ----------------
__global__ __launch_bounds__(256) void min_partial_kernel(const float* __restrict__ x,
                                                          float* __restrict__ partial, int n4) {
    __shared__ float smem[256];
    const float4* x4 = (const float4*)x;
    float m = 3.4028235e38f;
    for (int i = blockIdx.x * 256 + threadIdx.x; i < n4; i += gridDim.x * 256) {
        float4 v = x4[i];
        m = fminf(m, fminf(fminf(v.x, v.y), fminf(v.z, v.w)));
    }
    smem[threadIdx.x] = m;
    __syncthreads();
    for (int off = 128; off > 0; off >>= 1) {
        if (threadIdx.x < off)
            smem[threadIdx.x] = fminf(smem[threadIdx.x], smem[threadIdx.x + off]);
        __syncthreads();
    }
    if (threadIdx.x == 0) partial[blockIdx.x] = smem[0];
}

// ---------------- Pass 2: final min (padded zeros participate => min with 0) ----------------
__global__ __launch_bounds__(256) void min_final_kernel(const float* __restrict__ partial,
                                                        float* __restrict__ gmin) {
    __shared__ float smem[256];
    smem[threadIdx.x] = partial[threadIdx.x];
    __syncthreads();
    for (int off = 128; off > 0; off >>= 1) {
        if (threadIdx.x < off)
            smem[threadIdx.x] = fminf(smem[threadIdx.x], smem[threadIdx.x + off]);
        __syncthreads();
    }
    if (threadIdx.x == 0) gmin[0] = fminf(smem[0], 0.0f);
}

// ---------------- Pass 3: stencil + interleave ----------------
// One workgroup per (b, h), h in [0,63). Async-DMA rows h,h+1 into LDS; async-store the
// copy row(s) straight from LDS; compute p = w*s(w), q = s(w) once per input element
// (s = (w!=0)*2/(1+exp(w-gmin))); then each output pixel is a 6-tap linear combo of p,q.
__global__ __launch_bounds__(256) void upsample_kernel(const float* __restrict__ x,
                                                       float* __restrict__ out,
                                                       const float* __restrict__ gmin_ptr) {
    __shared__ float xrow[2 * WW];               // raw rows (16B-aligned chunks)
    __shared__ float P[2 * PQSTRIDE];            // p = w * s(w), with halos
    __shared__ float Q[2 * PQSTRIDE];            // q = s(w), with halos
    const int blk = blockIdx.x;
    const int b = blk / (HH - 1);
    const int h = blk % (HH - 1);
    const int tid = threadIdx.x;
    const float gmin = gmin_ptr[0];

    const float* row0 = x + (size_t)(b * HH + h) * WW;
    float* outb = out + (size_t)b * (2 * HH) * WW;

    // zero halos of the p/q arrays (padding value 0 -> mask 0 -> p=q=0)
    if (tid < 8) {
        int r = tid & 1;
        int e = (tid >> 1) & 1;          // 0: left, 1: right
        int idx = r * PQSTRIDE + (e ? (1 + WW) : 0);
        if (tid < 4) P[idx] = 0.0f; else Q[idx] = 0.0f;
    }

    // Stage 2 rows x 512 float4 chunks via async DMA: 4 chunks per thread.
#if HAVE_ASYNC_LOAD
    #pragma unroll
    for (int k = 0; k < 4; ++k) {
        int q  = tid + 256 * k;         // 0..1023
        int r  = q >> 9;                // row 0/1
        int c4 = q & 511;               // float4 index within row
        const float* gsrc = row0 + (size_t)r * WW + c4 * 4;
        float* ldst = &xrow[r * WW + c4 * 4];
        __builtin_amdgcn_global_load_async_to_lds_b128((v4i*)const_cast<float*>(gsrc),
                                                       (v4i*)ldst, 0, 0);
    }
    wait_async0();
#else
    #pragma unroll
    for (int k = 0; k < 4; ++k) {
        int q  = tid + 256 * k;
        int r  = q >> 9;
        int c4 = q & 511;
        float4 v = *(const float4*)(row0 + (size_t)r * WW + c4 * 4);
        *(float4*)&xrow[r * WW + c4 * 4] = v;
    }
#endif
    __syncthreads();

    // Copy rows straight out of LDS via async DMA (overlaps with the math below).
#if HAVE_ASYNC_STORE
    #pragma unroll
    for (int k = 0; k < 2; ++k) {
        int c4 = tid + 256 * k;
        __builtin_amdgcn_global_store_async_from_lds_b128(
            (v4i*)(outb + (size_t)(2 * h) * WW + c4 * 4), (v4i*)&xrow[c4 * 4], 0, 0);
    }
    if (h == HH - 2) {
        #pragma unroll
        for (int k = 0; k < 2; ++k) {
            int c4 = tid + 256 * k;
            __builtin_amdgcn_global_store_async_from_lds_b128(
                (v4i*)(outb + (size_t)(2 * HH - 2) * WW + c4 * 4),
                (v4i*)&xrow[WW + c4 * 4], 0, 0);
            __builtin_amdgcn_global_store_async_from_lds_b128(
                (v4i*)(outb + (size_t)(2 * HH - 1) * WW + c4 * 4),
                (v4i*)&xrow[WW + c4 * 4], 0, 0);
        }
    }
#else
    #pragma unroll
    for (int k = 0; k < 2; ++k) {
        int c4 = tid + 256 * k;
        float4 v = *(const float4*)&xrow[c4 * 4];
        *(float4*)(outb + (size_t)(2 * h) * WW + c4 * 4) = v;
    }
    if (h == HH - 2) {
        #pragma unroll
        for (int k = 0; k < 2; ++k) {
            int c4 = tid + 256 * k;
            float4 v = *(const float4*)&xrow[WW + c4 * 4];
            *(float4*)(outb + (size_t)(2 * HH - 2) * WW + c4 * 4) = v;
            *(float4*)(outb + (size_t)(2 * HH - 1) * WW + c4 * 4) = v;
        }
    }
#endif

    // Per-element p/q: one v_exp per element (16 elements per thread).
    #pragma unroll
    for (int k = 0; k < 16; ++k) {
        int q  = tid + 256 * k;          // 0..4095
        int r  = q >> 11;                // row 0/1
        int c  = q & (WW - 1);           // column
        float w = xrow[r * WW + c];
        float s = 2.0f / (1.0f + __expf(w - gmin));
        s = (w != 0.0f) ? s : 0.0f;
        P[r * PQSTRIDE + 1 + c] = w * s;
        Q[r * PQSTRIDE + 1 + c] = s;
    }
    __syncthreads();

    // 6-tap linear combination per output pixel.
    const float ES = 0.49306869139523984f;   // exp(-0.5*sqrt(2))
    const float E1 = 0.60653065971263342f;   // exp(-0.5)
    float* orow = outb + (size_t)(2 * h + 1) * WW;
    #pragma unroll
    for (int k = 0; k < 8; ++k) {
        int c = tid + 256 * k;
        float pside = (P[c] + P[c + 2]) + (P[PQSTRIDE + c] + P[PQSTRIDE + c + 2]);
        float pmid  = P[c + 1] + P[PQSTRIDE + c + 1];
        float qside = (Q[c] + Q[c + 2]) + (Q[PQSTRIDE + c] + Q[PQSTRIDE + c + 2]);
        float qmid  = Q[c + 1] + Q[PQSTRIDE + c + 1];
        float num = ES * pside + E1 * pmid;
        float den = ES * qside + E1 * qmid;
        den = (den == 0.0f) ? 1.0f : den;
        orow[c] = num / den;
    }
}

extern "C" void kernel_launch(void* const* d_in, const int* in_sizes, int n_in,
                              void* d_out, int out_size, void* d_ws, size_t ws_size,
                              hipStream_t stream) {
    const float* x = (const float*)d_in[0];
    float* out = (float*)d_out;
    float* partial = (float*)d_ws;        // 256 floats
    float* gmin = partial + 256;          // 1 float
    int n = in_sizes[0];                  // 32*1*64*2048 = 4194304

    min_partial_kernel<<<256, 256, 0, stream>>>(x, partial, n / 4);
    min_final_kernel<<<1, 256, 0, stream>>>(partial, gmin);
    upsample_kernel<<<NB * (HH - 1), 256, 0, stream>>>(x, out, gmin);
}